// KBpA_19069654794375
// MI455X (gfx1250) — compile-verified
//
#include <hip/hip_runtime.h>

// ---------------------------------------------------------------------------
// Stage 1: deformed = pixels - K @ betas          (HBM-bound: 256 MB of K)
// Stage 2: out[p]   = sum_c exp2(S2*big[p,c]) * alpha[c]
//          big[p,c] = p2 + c2 - 2*dot  ==  rank-4 product -> V_WMMA_F32_16X16X4_F32
//          S2 = -log2(e)/(2*SDP2) folded into the A fragment so each element
//          costs exactly one v_exp_f32 + one fma.
// ---------------------------------------------------------------------------

typedef __attribute__((ext_vector_type(2))) float v2f;
typedef __attribute__((ext_vector_type(8))) float v8f;

#define SDP2_F 0.05f
#define LOG2E_F 1.4426950408889634f

// ---------------- Kernel 1: stream K, compute deformed pixels ---------------
__global__ __launch_bounds__(256) void deform_kernel(
    const float* __restrict__ Kmat, const float* __restrict__ betas,
    const float* __restrict__ pixels, float* __restrict__ deformed, int KB) {
  __shared__ float sb[2048];                 // betas interleaved [k][2]
  for (int i = threadIdx.x; i < 2 * KB; i += 256) sb[i] = betas[i];
  __syncthreads();

  int p = blockIdx.x * 256 + threadIdx.x;
  const float4* Krow = (const float4*)(Kmat + (size_t)p * KB);
  float ax = 0.f, ay = 0.f;
  int nlines = KB >> 5;                      // 128B cachelines per row (8 x float4)
  for (int line = 0; line < nlines; ++line) {
    // pull the stream ~1KB ahead (global_prefetch_b8)
    __builtin_prefetch(Krow + line * 8 + 64, 0, 0);
#pragma unroll
    for (int j = 0; j < 8; ++j) {
      int k4 = line * 8 + j;
      float4 kv = Krow[k4];
      const float* b = &sb[k4 * 8];          // uniform addr -> LDS broadcast
      ax = fmaf(kv.x, b[0], ax); ay = fmaf(kv.x, b[1], ay);
      ax = fmaf(kv.y, b[2], ax); ay = fmaf(kv.y, b[3], ay);
      ax = fmaf(kv.z, b[4], ax); ay = fmaf(kv.z, b[5], ay);
      ax = fmaf(kv.w, b[6], ax); ay = fmaf(kv.w, b[7], ay);
    }
  }
  float2 px = ((const float2*)pixels)[p];
  float2 d;
  d.x = px.x - ax;
  d.y = px.y - ay;
  ((float2*)deformed)[p] = d;
}

// ---------------- Kernel 2: fused RBF matrix via WMMA + exp + matvec --------
__global__ __launch_bounds__(256) void rbf_wmma_kernel(
    const float* __restrict__ deformed, const float* __restrict__ centers,
    const float* __restrict__ alphas, float* __restrict__ out, int C) {
  const float S2 = -LOG2E_F / (2.0f * SDP2_F);   // exp(-big/2s^2) = exp2(S2*big)

  __shared__ float sc[4096];                 // centers interleaved [c][2] (16KB)
  __shared__ float sa[2048];                 // alphas (8KB)
  for (int i = threadIdx.x; i < 2 * C; i += 256) sc[i] = centers[i];
  for (int i = threadIdx.x; i < C; i += 256) sa[i] = alphas[i];
  __syncthreads();

  int lane  = threadIdx.x & 31;
  int wave  = threadIdx.x >> 5;
  int ptile = blockIdx.x * 8 + wave;         // one wave = 16 pixels
  int prow  = lane & 15;
  bool lo   = lane < 16;

  // A fragment (16x4 f32, ISA layout): lanes 0-15 hold (K0,K1)=(s*dx, s*dy),
  // lanes 16-31 hold (K2,K3)=(s*p2, s).
  float2 dp = ((const float2*)deformed)[ptile * 16 + prow];
  float p2 = dp.x * dp.x + dp.y * dp.y;
  v2f A;
  A.x = S2 * (lo ? dp.x : p2);
  A.y = S2 * (lo ? dp.y : 1.0f);

  v8f acc = {};                              // per-lane alpha-weighted sums
  int ntiles = C >> 4;
  for (int t = 0; t < ntiles; ++t) {
    int c = t * 16 + prow;
    float cx = sc[2 * c], cy = sc[2 * c + 1];
    float c2 = cx * cx + cy * cy;
    // B fragment (4x16): lanes 0-15 -> (K0,K1)=(-2cx,-2cy); lanes 16-31 -> (1,c2)
    v2f B;
    B.x = lo ? (-2.0f * cx) : 1.0f;
    B.y = lo ? (-2.0f * cy) : c2;
    float alpha = sa[c];

    v8f z = {};
    v8f d = __builtin_amdgcn_wmma_f32_16x16x4_f32(
        false, A, false, B, (short)0, z, false, false);

    // kBp = exp2(S2*big); accumulate kBp * alpha  (alpha depends on N = lane%16)
    acc[0] += __builtin_amdgcn_exp2f(d[0]) * alpha;
    acc[1] += __builtin_amdgcn_exp2f(d[1]) * alpha;
    acc[2] += __builtin_amdgcn_exp2f(d[2]) * alpha;
    acc[3] += __builtin_amdgcn_exp2f(d[3]) * alpha;
    acc[4] += __builtin_amdgcn_exp2f(d[4]) * alpha;
    acc[5] += __builtin_amdgcn_exp2f(d[5]) * alpha;
    acc[6] += __builtin_amdgcn_exp2f(d[6]) * alpha;
    acc[7] += __builtin_amdgcn_exp2f(d[7]) * alpha;
  }

  // Reduce over N (the 16 lanes of each half-wave). Masks < 16 stay in-half.
#pragma unroll
  for (int m = 1; m < 16; m <<= 1) {
#pragma unroll
    for (int r = 0; r < 8; ++r) acc[r] += __shfl_xor(acc[r], m, 32);
  }

  // D layout: VGPR r, lanes 0-15 -> pixel M=r; lanes 16-31 -> pixel M=8+r.
  int base = ptile * 16 + ((lane >> 4) << 3);
#pragma unroll
  for (int r = 0; r < 8; ++r)
    if (prow == r) out[base + r] = acc[r];
}

// ---------------------------------------------------------------------------
extern "C" void kernel_launch(void* const* d_in, const int* in_sizes, int n_in,
                              void* d_out, int out_size, void* d_ws, size_t ws_size,
                              hipStream_t stream) {
  const float* alphas  = (const float*)d_in[0];   // [C,1]
  const float* betas   = (const float*)d_in[1];   // [KB,2]
  const float* Kmat    = (const float*)d_in[2];   // [P,KB]
  const float* pixels  = (const float*)d_in[3];   // [P,2]
  const float* centers = (const float*)d_in[4];   // [C,2]

  int P  = in_sizes[3] / 2;   // 65536
  int C  = in_sizes[4] / 2;   // 2048
  int KB = in_sizes[1] / 2;   // 1024

  float* deformed = (float*)d_ws;                 // P*2 floats = 512KB scratch

  deform_kernel<<<P / 256, 256, 0, stream>>>(Kmat, betas, pixels, deformed, KB);

  // one wave per 16-pixel tile, 8 waves per block
  rbf_wmma_kernel<<<(P / 16) / 8, 256, 0, stream>>>(deformed, centers, alphas,
                                                    (float*)d_out, C);
}